// GNNEnocder_25426206392672
// MI455X (gfx1250) — compile-verified
//
#include <hip/hip_runtime.h>
#include <hip/hip_bf16.h>
#include <math.h>

// GNN encoder: 3x (GCNConv [+ PairNorm + skip + GELU]) on N=10000, D=128, E=640000.
// GEMMs: fp32 WMMA (v_wmma_f32_16x16x4_f32), W staged into LDS via TDM
// (tensor_load_to_lds + s_wait_tensorcnt), 64x128 block tile, 4 tiles/wave.
// Edge aggregation: one wave32 per edge, float4 per lane, global f32 atomics.

#define DIM 128

typedef __attribute__((ext_vector_type(2))) float v2f;
typedef __attribute__((ext_vector_type(8))) float v8f;
typedef __attribute__((ext_vector_type(4))) unsigned v4u;
typedef __attribute__((ext_vector_type(8))) int v8i;
typedef __attribute__((ext_vector_type(4))) int v4i;

// ---------------- degree kernels ----------------
__global__ void deg_init_kernel(float* __restrict__ deg, int n) {
    int i = blockIdx.x * blockDim.x + threadIdx.x;
    if (i < n) deg[i] = 1.0f;  // self loop contributes 1
}

__global__ void deg_accum_kernel(float* __restrict__ deg,
                                 const int* __restrict__ dst, int e) {
    int i = blockIdx.x * blockDim.x + threadIdx.x;
    if (i < e) atomicAdd(&deg[dst[i]], 1.0f);
}

__global__ void deg_finalize_kernel(float* __restrict__ deg, int n) {
    int i = blockIdx.x * blockDim.x + threadIdx.x;
    if (i < n) deg[i] = rsqrtf(deg[i]);  // deg >= 1 always (self loop)
}

// ---------------- fp32 WMMA GEMM: H[N,128] = X[N,128] @ W[128,128] ----------------
// Block = 8 waves = 64 rows x 128 cols. W (64KB) staged into LDS once per block
// with the Tensor Data Mover; each wave computes 4 16x16 tiles sharing one
// A-fragment per K-step (4 WMMAs per A-load, B-fragments via ds_load).
__global__ void __launch_bounds__(256) gemm_wmma_kernel(const float* __restrict__ X,
                                                        const float* __restrict__ W,
                                                        float* __restrict__ H,
                                                        int nrows) {
    __shared__ float Wlds[DIM * DIM];  // 64 KB of the 320 KB WGP LDS

    // --- TDM: DMA the whole 128x128 f32 weight tile Global -> LDS (wave 0 only) ---
    if (threadIdx.x < 32) {
        unsigned lds_off = (unsigned)(size_t)(&Wlds[0]);     // LDS aperture low bits = byte offset
        unsigned long long ga = (unsigned long long)(size_t)W;
        v4u g0;  // D# group 0: count=1 | lds_addr | global_addr[56:0] | type=2
        g0[0] = 1u;                                          // count=1, is_restore=0, gather off
        g0[1] = lds_off;                                     // lds_addr (bytes)
        g0[2] = (unsigned)ga;                                // global_addr[31:0]
        g0[3] = (unsigned)(ga >> 32) | (2u << 30);           // global_addr[56:32] | type=2
        v8i g1;  // D# group 1
        g1[0] = 2 << 16;          // workgroup_mask=0, data_size=2 (4 bytes)
        g1[1] = DIM << 16;        // tensor_dim0[15:0]=128 (bits 79:48)
        g1[2] = DIM << 16;        // tensor_dim0[31:16]=0, tensor_dim1[15:0]=128
        g1[3] = DIM << 16;        // tensor_dim1[31:16]=0, tile_dim0=128
        g1[4] = DIM;              // tile_dim1=128, tile_dim2=0
        g1[5] = DIM;              // tensor_dim0_stride[31:0]=128
        g1[6] = 0;                // stride0 hi, tensor_dim1_stride lo (unused, 2D tile)
        g1[7] = 0;
        v4i gz = {0, 0, 0, 0};    // groups 2/3 unused (2D tensor)
#if defined(__clang_major__) && __clang_major__ >= 23
        v8i gz8 = {0, 0, 0, 0, 0, 0, 0, 0};
        __builtin_amdgcn_tensor_load_to_lds(g0, g1, gz, gz, gz8, 0);
#else
        __builtin_amdgcn_tensor_load_to_lds(g0, g1, gz, gz, 0);
#endif
        __builtin_amdgcn_s_wait_tensorcnt(0);
    }
    __syncthreads();

    int wave = threadIdx.x >> 5;
    int mt   = wave >> 1;                 // 0..3 -> 16-row strip in the 64-row block
    int nh   = wave & 1;                  // 0/1 -> 64-col half
    int lane = threadIdx.x & 31;
    int half = lane >> 4;                 // lane half (K split per WMMA fp32 A layout)
    int l    = lane & 15;

    int rowbase = blockIdx.x * 64 + mt * 16;
    int row  = rowbase + l;
    int rowc = row < nrows ? row : (nrows - 1);   // clamp loads; EXEC stays all-ones
    const float2* Xrow = (const float2*)(X + (size_t)rowc * DIM);

    v8f acc0 = {}, acc1 = {}, acc2 = {}, acc3 = {};
    int c0 = nh * 64 + l;                 // B-fragment column for tile 0 of this wave

    #pragma unroll 4
    for (int k = 0; k < DIM; k += 4) {
        int ka = k + half * 2;            // lanes 0-15: K=k,k+1 ; lanes 16-31: K=k+2,k+3
        float2 af = Xrow[ka >> 1];
        v2f a; a[0] = af.x; a[1] = af.y;
        const float* w0 = &Wlds[ka * DIM];
        const float* w1 = &Wlds[(ka + 1) * DIM];
        v2f b0, b1, b2, b3;
        b0[0] = w0[c0];      b0[1] = w1[c0];
        b1[0] = w0[c0 + 16]; b1[1] = w1[c0 + 16];
        b2[0] = w0[c0 + 32]; b2[1] = w1[c0 + 32];
        b3[0] = w0[c0 + 48]; b3[1] = w1[c0 + 48];
        acc0 = __builtin_amdgcn_wmma_f32_16x16x4_f32(false, a, false, b0, (short)0, acc0, false, false);
        acc1 = __builtin_amdgcn_wmma_f32_16x16x4_f32(false, a, false, b1, (short)0, acc1, false, false);
        acc2 = __builtin_amdgcn_wmma_f32_16x16x4_f32(false, a, false, b2, (short)0, acc2, false, false);
        acc3 = __builtin_amdgcn_wmma_f32_16x16x4_f32(false, a, false, b3, (short)0, acc3, false, false);
    }

    if (rowbase < nrows) {  // nrows % 16 == 0 -> whole strip valid (block-strip uniform)
        int n0 = nh * 64;
        #pragma unroll
        for (int v = 0; v < 8; ++v) {
            int r = rowbase + half * 8 + v;
            float* o = H + (size_t)r * DIM + n0 + l;
            o[0]  = acc0[v];
            o[16] = acc1[v];
            o[32] = acc2[v];
            o[48] = acc3[v];
        }
    }
}

// ---------------- out = bias + self-loop term ----------------
__global__ void conv_init_kernel(float* __restrict__ out,
                                 const float* __restrict__ bias,
                                 const float* __restrict__ dis,
                                 const float* __restrict__ h, int total) {
    int idx = blockIdx.x * blockDim.x + threadIdx.x;
    if (idx >= total) return;
    int i = idx >> 7;
    int d = idx & (DIM - 1);
    float w = dis[i];
    out[idx] = bias[d] + w * w * h[idx];
}

// ---------------- edge scatter: one wave per edge, float4 per lane ----------------
__global__ void __launch_bounds__(256) edge_scatter_kernel(
        const float* __restrict__ h, const float* __restrict__ dis,
        const int* __restrict__ src, const int* __restrict__ dst,
        float* __restrict__ out, int e) {
    int edge = blockIdx.x * (blockDim.x >> 5) + (threadIdx.x >> 5);
    if (edge >= e) return;
    int lane = threadIdx.x & 31;
    int s = src[edge];
    int t = dst[edge];
    float nrm = dis[s] * dis[t];
    const float4* hs = (const float4*)(h + (size_t)s * DIM);
    float4 v = hs[lane];
    float* o = out + (size_t)t * DIM + lane * 4;
    atomicAdd(o + 0, nrm * v.x);
    atomicAdd(o + 1, nrm * v.y);
    atomicAdd(o + 2, nrm * v.z);
    atomicAdd(o + 3, nrm * v.w);
}

// ---------------- fused PairNorm('PN') + skip + exact GELU ----------------
__global__ void __launch_bounds__(256) fuse_pn_kernel(
        const float* __restrict__ agg, const float* __restrict__ prev_in,
        float* __restrict__ prev_out, float* __restrict__ x_out, int n) {
    int node = blockIdx.x * (blockDim.x >> 5) + (threadIdx.x >> 5);
    if (node >= n) return;
    int lane = threadIdx.x & 31;
    const float4* a = (const float4*)(agg + (size_t)node * DIM);
    float4 v = a[lane];
    float ss = v.x * v.x + v.y * v.y + v.z * v.z + v.w * v.w;
    #pragma unroll
    for (int off = 16; off > 0; off >>= 1)
        ss += __shfl_xor(ss, off, 32);
    float inv = 1.0f / (sqrtf(ss) + 1e-8f);
    const float4* p = (const float4*)(prev_in + (size_t)node * DIM);
    float4 pv = p[lane];
    float4 y;
    y.x = v.x * inv + pv.x;
    y.y = v.y * inv + pv.y;
    y.z = v.z * inv + pv.z;
    y.w = v.w * inv + pv.w;
    ((float4*)(prev_out + (size_t)node * DIM))[lane] = y;
    const float k = 0.70710678118654752f;  // 1/sqrt(2)
    float4 g;
    g.x = 0.5f * y.x * (1.0f + erff(y.x * k));
    g.y = 0.5f * y.y * (1.0f + erff(y.y * k));
    g.z = 0.5f * y.z * (1.0f + erff(y.z * k));
    g.w = 0.5f * y.w * (1.0f + erff(y.w * k));
    ((float4*)(x_out + (size_t)node * DIM))[lane] = g;
}

extern "C" void kernel_launch(void* const* d_in, const int* in_sizes, int n_in,
                              void* d_out, int out_size, void* d_ws, size_t ws_size,
                              hipStream_t stream) {
    const float* x  = (const float*)d_in[0];
    const int*   ei = (const int*)d_in[1];   // [2, E] row-major
    const float* W1 = (const float*)d_in[2];
    const float* b1 = (const float*)d_in[3];
    const float* W2 = (const float*)d_in[4];
    const float* b2 = (const float*)d_in[5];
    const float* W3 = (const float*)d_in[6];
    const float* b3 = (const float*)d_in[7];

    const int N = in_sizes[0] / DIM;
    const int E = in_sizes[1] / 2;
    const int* src = ei;
    const int* dst = ei + E;

    // workspace layout (floats): dis[N] | h[N*D] | agg[N*D] | prev[N*D] | xbuf[N*D]
    float* dis  = (float*)d_ws;
    float* h    = dis + N;
    float* agg  = h + (size_t)N * DIM;
    float* prev = agg + (size_t)N * DIM;
    float* xbuf = prev + (size_t)N * DIM;

    const int tpb = 256;
    const int ndBlocks  = (N * DIM + tpb - 1) / tpb;
    const int nBlocks   = (N + tpb - 1) / tpb;
    const int eBlocks   = (E + tpb - 1) / tpb;
    const int mBlocks   = (N + 63) / 64;      // 64 rows per block
    const int edgeWaves = (E + 7) / 8;        // 8 waves/block, 1 edge/wave
    const int nodeWaves = (N + 7) / 8;        // 8 waves/block, 1 node/wave

    // --- degrees ---
    deg_init_kernel<<<nBlocks, tpb, 0, stream>>>(dis, N);
    deg_accum_kernel<<<eBlocks, tpb, 0, stream>>>(dis, dst, E);
    deg_finalize_kernel<<<nBlocks, tpb, 0, stream>>>(dis, N);

    // --- layer 1: conv(x, W1, b1) -> pairnorm -> +x -> gelu ---
    gemm_wmma_kernel<<<mBlocks, tpb, 0, stream>>>(x, W1, h, N);
    conv_init_kernel<<<ndBlocks, tpb, 0, stream>>>(agg, b1, dis, h, N * DIM);
    edge_scatter_kernel<<<edgeWaves, tpb, 0, stream>>>(h, dis, src, dst, agg, E);
    fuse_pn_kernel<<<nodeWaves, tpb, 0, stream>>>(agg, x, prev, xbuf, N);

    // --- layer 2 ---
    gemm_wmma_kernel<<<mBlocks, tpb, 0, stream>>>(xbuf, W2, h, N);
    conv_init_kernel<<<ndBlocks, tpb, 0, stream>>>(agg, b2, dis, h, N * DIM);
    edge_scatter_kernel<<<edgeWaves, tpb, 0, stream>>>(h, dis, src, dst, agg, E);
    fuse_pn_kernel<<<nodeWaves, tpb, 0, stream>>>(agg, prev, prev, xbuf, N);

    // --- layer 3: conv only, straight into d_out ---
    float* out = (float*)d_out;
    gemm_wmma_kernel<<<mBlocks, tpb, 0, stream>>>(xbuf, W3, h, N);
    conv_init_kernel<<<ndBlocks, tpb, 0, stream>>>(out, b3, dis, h, N * DIM);
    edge_scatter_kernel<<<edgeWaves, tpb, 0, stream>>>(h, dis, src, dst, out, E);
}